// GeometricEncoder_85787676770444
// MI455X (gfx1250) — compile-verified
//
#include <hip/hip_runtime.h>

typedef __attribute__((ext_vector_type(2))) float v2f;
typedef __attribute__((ext_vector_type(8))) float v8f;
typedef __attribute__((ext_vector_type(4))) float f4v;
typedef int v4i_ __attribute__((vector_size(16)));  // matches builtin's pointee type

#define TPB 256
#define TILE_PTS 1024  // points staged per block: 1024 * 12B = 12288 B of LDS

// ---------------- async copy helpers (CDNA5 ASYNCcnt path) ----------------
#if defined(__HIP_DEVICE_COMPILE__) && __has_builtin(__builtin_amdgcn_global_load_async_to_lds_b128)
#define HAVE_ASYNC_LDS 1
__device__ __forceinline__ void async_cp16(const float* g, float* l) {
  // 16 bytes per lane: global -> LDS, tracked by ASYNCcnt
  __builtin_amdgcn_global_load_async_to_lds_b128(
      (__attribute__((address_space(1))) v4i_*)(unsigned long long)(g),
      (__attribute__((address_space(3))) v4i_*)(unsigned int)(unsigned long long)(l),
      0, 0);
}
#else
#define HAVE_ASYNC_LDS 0
__device__ __forceinline__ void async_cp16(const float* g, float* l) {
  *(f4v*)l = *(const f4v*)g;  // synchronous fallback
}
#endif

__device__ __forceinline__ void async_wait0() {
#if defined(__HIP_DEVICE_COMPILE__) && __has_builtin(__builtin_amdgcn_s_wait_asynccnt)
  __builtin_amdgcn_s_wait_asynccnt(0);
#elif defined(__HIP_DEVICE_COMPILE__) && defined(__AMDGCN__)
  asm volatile("s_wait_asynccnt 0" ::: "memory");
#endif
}

// ---------------- fp32 HW atomic (lands in L2; accumulators stay L2-resident) ----------------
__device__ __forceinline__ void atomAddF(float* p, float v) {
#if defined(__HIP_DEVICE_COMPILE__)
  unsafeAtomicAdd(p, v);
#else
  (void)p; (void)v;
#endif
}

__device__ __forceinline__ void accum_point(float* __restrict__ mom, int c,
                                            float x, float y, float z) {
  float* a = mom + (size_t)c * 10;
  atomAddF(a + 0, 1.0f);
  atomAddF(a + 1, x);
  atomAddF(a + 2, y);
  atomAddF(a + 3, z);
  atomAddF(a + 4, x * x);
  atomAddF(a + 5, x * y);
  atomAddF(a + 6, x * z);
  atomAddF(a + 7, y * y);
  atomAddF(a + 8, y * z);
  atomAddF(a + 9, z * z);
}

// ---------------- stage 2: raw-moment accumulation ----------------
__global__ __launch_bounds__(TPB) void k_accum(const float* __restrict__ coords,
                                               const int* __restrict__ cid,
                                               float* __restrict__ mom, int Npts) {
  __shared__ float tile[TILE_PTS * 3];  // 3072 floats = 12288 B
  const int t = threadIdx.x;
  const long long base = (long long)blockIdx.x * TILE_PTS;
  const bool full = (base + TILE_PTS) <= (long long)Npts;

  if (full) {
    const float* g = coords + base * 3;
    // 1024 pts * 12B = 12288B per block; 256 threads * 3 chunks * 16B
    #pragma unroll
    for (int k2 = 0; k2 < 3; ++k2) {
      const int fo = t * 4 + k2 * 1024;  // float offset, 16B-aligned
      async_cp16(g + fo, tile + fo);
    }
    async_wait0();
  }
  __syncthreads();

  if (full) {
    const int4 cc = *(const int4*)(cid + base + (long long)t * 4);
    const int cs[4] = {cc.x, cc.y, cc.z, cc.w};
    #pragma unroll
    for (int i = 0; i < 4; ++i) {
      const int l = t * 4 + i;
      accum_point(mom, cs[i], tile[l * 3 + 0], tile[l * 3 + 1], tile[l * 3 + 2]);
    }
  } else {
    for (int i = 0; i < 4; ++i) {
      const long long p = base + (long long)t * 4 + i;
      if (p < (long long)Npts) {
        accum_point(mom, cid[p], coords[p * 3 + 0], coords[p * 3 + 1], coords[p * 3 + 2]);
      }
    }
  }
}

// ---------------- stage 3: per-cluster eigensolve (one wave / cluster, WMMA for PxQ) ----------------
__global__ __launch_bounds__(128) void k_eigen(const float* __restrict__ mom,
                                               float* __restrict__ out,
                                               float* __restrict__ wsV,
                                               float* __restrict__ wsC,
                                               float* __restrict__ wsD, int K) {
  const int lane = threadIdx.x & 31;
  const int wv = threadIdx.x >> 5;
  const int c = blockIdx.x * 4 + wv;
  if (c >= K) return;  // wave-uniform; active waves keep EXEC all-ones for WMMA

  const float* m = mom + (size_t)c * 10;
  const float n = m[0];
  const float Sx = m[1], Sy = m[2], Sz = m[3];
  const float rn = n > 0.f ? 1.f / n : 0.f;
  const float cx = Sx * rn, cy = Sy * rn, cz = Sz * rn;
  const float a00 = m[4] - Sx * cx;
  const float a01 = m[5] - Sx * cy;
  const float a02 = m[6] - Sx * cz;
  const float a11 = m[7] - Sy * cy;
  const float a12 = m[8] - Sy * cz;
  const float a22 = m[9] - Sz * cz;

  // closed-form symmetric 3x3 eigenvalues (ascending w0 <= w1 <= w2)
  const float q = (a00 + a11 + a22) * (1.f / 3.f);
  const float d0 = a00 - q, d1 = a11 - q, d2 = a22 - q;
  const float p1 = a01 * a01 + a02 * a02 + a12 * a12;
  const float p2 = d0 * d0 + d1 * d1 + d2 * d2 + 2.f * p1;
  const float p = sqrtf(p2 * (1.f / 6.f));
  float w0, w1, w2;
  if (p > 1e-30f) {
    const float ip = 1.f / p;
    const float b00 = d0 * ip, b11 = d1 * ip, b22 = d2 * ip;
    const float b01 = a01 * ip, b02 = a02 * ip, b12 = a12 * ip;
    const float detB = b00 * (b11 * b22 - b12 * b12)
                     - b01 * (b01 * b22 - b12 * b02)
                     + b02 * (b01 * b12 - b11 * b02);
    const float r = fminf(1.f, fmaxf(-1.f, 0.5f * detB));
    const float phi = acosf(r) * (1.f / 3.f);
    w2 = q + 2.f * p * cosf(phi);
    w0 = q + 2.f * p * cosf(phi + 2.0943951023931953f);
    w1 = 3.f * q - w0 - w2;
  } else {
    w0 = w1 = w2 = q;
  }

  // V = (A - w0 I)(A - w1 I) = scale * v2 v2^T  -> any nonzero column is the principal axis.
  const float P00 = a00 - w0, P11 = a11 - w0, P22 = a22 - w0;
  const float Q00 = a00 - w1, Q11 = a11 - w1, Q22 = a22 - w1;

  const int half = lane >> 4;
  const int sub = lane & 15;
  // row `sub` of P (zero for sub >= 3)
  const float Pm0 = sub == 0 ? P00 : sub == 1 ? a01 : sub == 2 ? a02 : 0.f;
  const float Pm1 = sub == 0 ? a01 : sub == 1 ? P11 : sub == 2 ? a12 : 0.f;
  const float Pm2 = sub == 0 ? a02 : sub == 1 ? a12 : sub == 2 ? P22 : 0.f;
  // row `sub` of Q == column `sub` of Q (symmetric)
  const float Qm0 = sub == 0 ? Q00 : sub == 1 ? a01 : sub == 2 ? a02 : 0.f;
  const float Qm1 = sub == 0 ? a01 : sub == 1 ? Q11 : sub == 2 ? a12 : 0.f;
  const float Qm2 = sub == 0 ? a02 : sub == 1 ? a12 : sub == 2 ? Q22 : 0.f;

  float V0, V1, V2;  // rows 0..2 of column `sub` of V (valid in lanes 0..2)
#if defined(__HIP_DEVICE_COMPILE__) && __has_builtin(__builtin_amdgcn_wmma_f32_16x16x4_f32)
  // A-operand 16x4 f32: lanes 0-15 hold k={0,1} in v[0],v[1]; lanes 16-31 hold k={2,3}
  v2f Aop;
  Aop.x = half ? Pm2 : Pm0;
  Aop.y = half ? 0.f : Pm1;
  // B-operand 4x16 f32: VGPR0 rows k={0 | 2}, VGPR1 rows k={1 | 3}, cols striped over lanes
  v2f Bop;
  Bop.x = half ? Qm2 : Qm0;
  Bop.y = half ? 0.f : Qm1;
  v8f D = {0.f, 0.f, 0.f, 0.f, 0.f, 0.f, 0.f, 0.f};
  D = __builtin_amdgcn_wmma_f32_16x16x4_f32(false, Aop, false, Bop, (short)0, D,
                                            false, false);
  V0 = D[0];  // M=0..7 rows live in lanes 0-15, VGPR r = row r
  V1 = D[1];
  V2 = D[2];
#else
  // scalar fallback: V[i][sub] = P(row i) . Q(row sub)
  V0 = P00 * Qm0 + a01 * Qm1 + a02 * Qm2;
  V1 = a01 * Qm0 + P11 * Qm1 + a12 * Qm2;
  V2 = a02 * Qm0 + a12 * Qm1 + P22 * Qm2;
#endif

  // pick the column with the largest norm, broadcast, normalize
  const float cn2 = V0 * V0 + V1 * V1 + V2 * V2;
  const float q0 = __shfl(cn2, 0, 32);
  const float q1 = __shfl(cn2, 1, 32);
  const float q2 = __shfl(cn2, 2, 32);
  const int j = (q0 >= q1 && q0 >= q2) ? 0 : (q1 >= q2 ? 1 : 2);
  float vx = __shfl(V0, j, 32);
  float vy = __shfl(V1, j, 32);
  float vz = __shfl(V2, j, 32);
  const float len2 = vx * vx + vy * vy + vz * vz;
  if (len2 > 1e-30f) {
    const float il = rsqrtf(len2);
    vx *= il; vy *= il; vz *= il;
  } else {
    vx = 1.f; vy = 0.f; vz = 0.f;
  }

  const float rw2 = (w2 != 0.f) ? 1.f / w2 : 0.f;
  const float dirwt = 1.f - w1 * rw2;

  if (lane == 0) {
    float* o = out + (size_t)c * 19;
    o[0] = cx; o[1] = cy; o[2] = cz;
    o[3] = a00 * rw2;  o[4] = a01 * rw2;  o[5] = a02 * rw2;
    o[6] = a01 * rw2;  o[7] = a11 * rw2;  o[8] = a12 * rw2;
    o[9] = a02 * rw2;  o[10] = a12 * rw2; o[11] = a22 * rw2;
    // o[12..14] (oriented v0) written by k_fin after the sign pass
    o[15] = w0; o[16] = w1; o[17] = w2; o[18] = n;
    wsV[c * 3 + 0] = vx; wsV[c * 3 + 1] = vy; wsV[c * 3 + 2] = vz;
    wsC[c * 3 + 0] = cx; wsC[c * 3 + 1] = cy; wsC[c * 3 + 2] = cz;
    wsD[c] = dirwt;
  }
}

// ---------------- stage 4: orientation sign pass ----------------
__global__ __launch_bounds__(TPB) void k_sign(const float* __restrict__ coords,
                                              const int* __restrict__ cid,
                                              const float* __restrict__ wsV,
                                              const float* __restrict__ wsC,
                                              float* __restrict__ sc, int Npts) {
  const int i = blockIdx.x * blockDim.x + threadIdx.x;
  if (i >= Npts) return;
  const int c = cid[i];
  const float xc = coords[3 * (size_t)i + 0] - wsC[c * 3 + 0];
  const float yc = coords[3 * (size_t)i + 1] - wsC[c * 3 + 1];
  const float zc = coords[3 * (size_t)i + 2] - wsC[c * 3 + 2];
  const float vx = wsV[c * 3 + 0], vy = wsV[c * 3 + 1], vz = wsV[c * 3 + 2];
  const float x0 = xc * vx + yc * vy + zc * vz;
  const float px = xc - x0 * vx;
  const float py = yc - x0 * vy;
  const float pz = zc - x0 * vz;
  const float np0 = sqrtf(px * px + py * py + pz * pz);
  atomAddF(&sc[c], x0 * np0);
}

// ---------------- stage 5: finalize oriented axis ----------------
__global__ __launch_bounds__(TPB) void k_fin(const float* __restrict__ sc,
                                             const float* __restrict__ wsV,
                                             const float* __restrict__ wsD,
                                             float* __restrict__ out, int K) {
  const int c = blockIdx.x * blockDim.x + threadIdx.x;
  if (c >= K) return;
  const float s = sc[c] < 0.f ? -1.f : 1.f;
  const float f = wsD[c] * s;
  float* o = out + (size_t)c * 19 + 12;
  o[0] = f * wsV[c * 3 + 0];
  o[1] = f * wsV[c * 3 + 1];
  o[2] = f * wsV[c * 3 + 2];
}

extern "C" void kernel_launch(void* const* d_in, const int* in_sizes, int n_in,
                              void* d_out, int out_size, void* d_ws, size_t ws_size,
                              hipStream_t stream) {
  const float* coords = (const float*)d_in[0];
  const int* cid = (const int*)d_in[1];
  const int Npts = in_sizes[0] / 3;
  const int K = out_size / 19;

  float* ws = (float*)d_ws;
  float* mom = ws;                        // K*10: n, Sx..Sz, Mxx,Mxy,Mxz,Myy,Myz,Mzz
  float* wsV = mom + (size_t)K * 10;      // K*3 : principal axis
  float* wsC = wsV + (size_t)K * 3;       // K*3 : center
  float* wsD = wsC + (size_t)K * 3;       // K   : dirwt
  float* sc  = wsD + K;                   // K   : orientation sum
  const size_t zero_bytes = sizeof(float) * (size_t)K * 18;

  (void)hipMemsetAsync(d_ws, 0, zero_bytes, stream);

  const int tiles = (Npts + TILE_PTS - 1) / TILE_PTS;
  k_accum<<<tiles, TPB, 0, stream>>>(coords, cid, mom, Npts);

  k_eigen<<<(K + 3) / 4, 128, 0, stream>>>(mom, (float*)d_out, wsV, wsC, wsD, K);

  k_sign<<<(Npts + TPB - 1) / TPB, TPB, 0, stream>>>(coords, cid, wsV, wsC, sc, Npts);

  k_fin<<<(K + TPB - 1) / TPB, TPB, 0, stream>>>(sc, wsV, wsD, (float*)d_out, K);
}